// CapsuleLayer_15444702396775
// MI455X (gfx1250) — compile-verified
//
#include <hip/hip_runtime.h>

typedef __attribute__((ext_vector_type(2))) float v2f;
typedef __attribute__((ext_vector_type(8))) float v8f;

#define B_  128
#define I_  1152
#define K_  8
#define O_  32
#define D_  16
#define NI  16                 // input caps per workgroup chunk
#define NCHUNK (I_ / NI)       // 72

// Fused routing iteration:
//  - recompute x_hat tiles (16 b-rows x 16 d-cols per capsule o) with v_wmma_f32_16x16x4_f32
//  - ITER>0: agreement a = <x_hat, v>, logit update, softmax over o -> c
//  - accumulate s_partial[b,o,d] += c * x_hat in WMMA accumulator layout
// One workgroup = 128 threads = 4 waves; wave w owns capsules o in [8w, 8w+8).
template <int ITER>
__global__ __launch_bounds__(128, 1)
void k_route(const float* __restrict__ x,     // (B, I, K)
             const float* __restrict__ W,     // (I, O, K, D)
             const float* __restrict__ vin,   // (B, O, D)   (ITER > 0)
             float* __restrict__ blog,        // (B, I, O)
             float* __restrict__ s_part)      // (NCHUNK, B, O, D)
{
    __shared__ float a_lds[16 * 32];   // a[m][o]
    __shared__ float c_lds[16 * 32];   // c[m][o]

    const int t    = threadIdx.x;
    const int lane = t & 31;
    const int wv   = t >> 5;          // wave id 0..3
    const int g    = lane >> 4;       // half-wave group 0/1
    const int ln   = lane & 15;
    const int b0   = blockIdx.x * 16; // batch tile
    const int i0   = blockIdx.y * NI; // input-capsule chunk

    v8f s_acc[8];
    #pragma unroll
    for (int ol = 0; ol < 8; ++ol)
        #pragma unroll
        for (int r = 0; r < 8; ++r) s_acc[ol][r] = 0.0f;

    for (int ii = 0; ii < NI; ++ii) {
        const int i = i0 + ii;

        // ---- A fragments: A[m, k] = x[b0+m, i, k]; lane holds k = 2g + elt ----
        const float* xrow = x + ((size_t)(b0 + ln) * I_ + i) * K_;
        const float2 xa = *reinterpret_cast<const float2*>(xrow + 2 * g);       // k0 = 0
        const float2 xb = *reinterpret_cast<const float2*>(xrow + 4 + 2 * g);   // k0 = 4
        v2f a0; a0.x = xa.x; a0.y = xa.y;
        v2f a1; a1.x = xb.x; a1.y = xb.y;

        // ---- x_hat tiles per owned capsule ----
        v8f xh[8];
        #pragma unroll
        for (int ol = 0; ol < 8; ++ol) {
            const int o = wv * 8 + ol;
            // B[k, d] = W[i, o, k, d]; lane holds (k = k0 + 2g + elt, d = ln)
            const float* wb = W + ((size_t)i * O_ + o) * (K_ * D_) + ln;
            v2f bf0; bf0.x = wb[(2 * g + 0) * D_]; bf0.y = wb[(2 * g + 1) * D_];
            v2f bf1; bf1.x = wb[(4 + 2 * g) * D_]; bf1.y = wb[(5 + 2 * g) * D_];
            v8f acc;
            #pragma unroll
            for (int r = 0; r < 8; ++r) acc[r] = 0.0f;
            acc = __builtin_amdgcn_wmma_f32_16x16x4_f32(false, a0, false, bf0,
                                                        (short)0, acc, false, false);
            acc = __builtin_amdgcn_wmma_f32_16x16x4_f32(false, a1, false, bf1,
                                                        (short)0, acc, false, false);
            xh[ol] = acc;
        }

        // ---- agreement: a[m, o] = sum_d x_hat[m, d] * v[b0+m, o, d] ----
        if constexpr (ITER > 0) {
            #pragma unroll
            for (int ol = 0; ol < 8; ++ol) {
                const int o = wv * 8 + ol;
                float ar[8];
                #pragma unroll
                for (int r = 0; r < 8; ++r) {
                    const float vv = vin[((size_t)(b0 + r + 8 * g) * O_ + o) * D_ + ln];
                    float p = xh[ol][r] * vv;
                    p += __shfl_xor(p, 1);
                    p += __shfl_xor(p, 2);
                    p += __shfl_xor(p, 4);
                    p += __shfl_xor(p, 8);   // stays inside 16-lane group
                    ar[r] = p;
                }
                if (ln == 0) {
                    #pragma unroll
                    for (int r = 0; r < 8; ++r)
                        a_lds[(r + 8 * g) * 32 + o] = ar[r];
                }
            }
        }
        __syncthreads();

        // ---- logit update + softmax over o (8 threads per batch row) ----
        if constexpr (ITER > 0) {
            const int m  = t >> 3;   // 0..15, lanes 8m..8m+7 in one wave
            const int oj = t & 7;
            float bn[4];
            #pragma unroll
            for (int j = 0; j < 4; ++j) {
                const int o = oj + 8 * j;
                const size_t bi = ((size_t)(b0 + m) * I_ + i) * O_ + o;
                float bo = (ITER >= 2) ? blog[bi] : 0.0f;
                bn[j] = bo + a_lds[m * 32 + o];
                if (ITER < 2) blog[bi] = bn[j];   // last iter: logits unused afterwards
            }
            float mx = fmaxf(fmaxf(bn[0], bn[1]), fmaxf(bn[2], bn[3]));
            mx = fmaxf(mx, __shfl_xor(mx, 1));
            mx = fmaxf(mx, __shfl_xor(mx, 2));
            mx = fmaxf(mx, __shfl_xor(mx, 4));
            float e[4], sum = 0.0f;
            #pragma unroll
            for (int j = 0; j < 4; ++j) { e[j] = __expf(bn[j] - mx); sum += e[j]; }
            sum += __shfl_xor(sum, 1);
            sum += __shfl_xor(sum, 2);
            sum += __shfl_xor(sum, 4);
            const float inv = 1.0f / sum;
            #pragma unroll
            for (int j = 0; j < 4; ++j)
                c_lds[m * 32 + oj + 8 * j] = e[j] * inv;
        }
        __syncthreads();

        // ---- s accumulation in accumulator layout: s[m, d] += c[m] * xh[m, d] ----
        #pragma unroll
        for (int ol = 0; ol < 8; ++ol) {
            const int o = wv * 8 + ol;
            #pragma unroll
            for (int r = 0; r < 8; ++r) {
                const float c = (ITER == 0) ? (1.0f / 32.0f)
                                            : c_lds[(r + 8 * g) * 32 + o];  // LDS broadcast
                s_acc[ol][r] += c * xh[ol][r];
            }
        }
        __syncthreads();
    }

    // ---- write partial sums: s_part[chunk, b, o, d] ----
    #pragma unroll
    for (int ol = 0; ol < 8; ++ol) {
        const int o = wv * 8 + ol;
        #pragma unroll
        for (int r = 0; r < 8; ++r)
            s_part[(((size_t)blockIdx.y * B_ + b0 + r + 8 * g) * O_ + o) * D_ + ln] =
                s_acc[ol][r];
    }
}

// Reduce partial sums over chunks, then squash each (b, o) row of length D=16.
__global__ __launch_bounds__(256)
void k_squash(const float* __restrict__ s_part, float* __restrict__ vout)
{
    const int t = blockIdx.x * blockDim.x + threadIdx.x;  // 0 .. B*O*D-1, d = t & 15
    float s = 0.0f;
    for (int c = 0; c < NCHUNK; ++c)
        s += s_part[(size_t)c * (B_ * O_ * D_) + t];
    float sq = s * s;
    sq += __shfl_xor(sq, 1);
    sq += __shfl_xor(sq, 2);
    sq += __shfl_xor(sq, 4);
    sq += __shfl_xor(sq, 8);   // sum over the 16 d-lanes of this row
    vout[t] = (sq / (1.0f + sq)) * s * rsqrtf(sq + 1e-8f);
}

extern "C" void kernel_launch(void* const* d_in, const int* in_sizes, int n_in,
                              void* d_out, int out_size, void* d_ws, size_t ws_size,
                              hipStream_t stream) {
    const float* x = (const float*)d_in[0];   // (128, 1152, 8)
    const float* W = (const float*)d_in[1];   // (1152, 32, 8, 16)
    float* out = (float*)d_out;               // (128, 32, 16)

    char* ws = (char*)d_ws;
    float* blog   = (float*)ws;                                           // B*I*O      (18.9 MB)
    float* s_part = (float*)(ws + (size_t)B_ * I_ * O_ * sizeof(float));  // NCHUNK*B*O*D (18.9 MB)
    float* vbuf   = (float*)(ws + (size_t)B_ * I_ * O_ * sizeof(float)
                                + (size_t)NCHUNK * B_ * O_ * D_ * sizeof(float)); // 256 KB

    dim3 grid(B_ / 16, NCHUNK), blk(128);
    const int nsq = (B_ * O_ * D_) / 256;

    // iter 1: c uniform (softmax of zero logits)
    k_route<0><<<grid, blk, 0, stream>>>(x, W, nullptr, blog, s_part);
    k_squash<<<nsq, 256, 0, stream>>>(s_part, vbuf);
    // iter 2: b = a1
    k_route<1><<<grid, blk, 0, stream>>>(x, W, vbuf, blog, s_part);
    k_squash<<<nsq, 256, 0, stream>>>(s_part, vbuf);
    // iter 3: b = a1 + a2
    k_route<2><<<grid, blk, 0, stream>>>(x, W, vbuf, blog, s_part);
    k_squash<<<nsq, 256, 0, stream>>>(s_part, out);
}